// PAGAT_37417755083713
// MI455X (gfx1250) — compile-verified
//
#include <hip/hip_runtime.h>
#include <hip/hip_bf16.h>

// MI455X / gfx1250, wave32. All matrix math via V_WMMA_F32_16X16X4_F32
// (fp32 A/B, fp32 accumulate) to match the fp32 reference exactly.

typedef float v2f __attribute__((ext_vector_type(2)));
typedef float v8f __attribute__((ext_vector_type(8)));

#define B_ 8
#define N_ 2048
#define F_ 128
#define NEG_ELU_ONE 1.0f

// ---------------------------------------------------------------------------
// Kernel 1: Wh[b,n,g] = sum_f h[b,n,f] * W[g,f]
// One wave per 16x16 output tile; K=128 via 32 chained 16x16x4 f32 WMMAs.
// A-frag (16x4 f32): lane L -> M = L%16, K-pair = 2*(L/16)  (contig float2)
// B-frag (4x16 f32): lane L -> N = L%16, K-pair = 2*(L/16); B[k][g] = W[g][k]
// C/D    (16x16 f32): vgpr v, lane L -> M = v + 8*(L/16), N = L%16
// ---------------------------------------------------------------------------
__global__ __launch_bounds__(256) void wh_gemm_kernel(
    const float* __restrict__ h, const float* __restrict__ W,
    float* __restrict__ Wh) {
  const int lane = threadIdx.x & 31;
  const int warp = threadIdx.x >> 5;
  const int ww   = blockIdx.x * 8 + warp;      // 8192 tiles total
  const int b    = ww >> 10;                   // (N/16)*(F/16) = 1024 per batch
  const int rem  = ww & 1023;
  const int mt   = rem >> 3;                   // 0..127 row tiles
  const int gt   = rem & 7;                    // 0..7  col tiles
  const int r    = lane & 15;
  const int kp   = (lane >> 4) << 1;           // 0 or 2

  const float* arow = h + ((size_t)(b * N_ + mt * 16 + r)) * F_ + kp;
  const float* brow = W + (size_t)(gt * 16 + r) * F_ + kp;

  v8f c = {};
#pragma unroll
  for (int k0 = 0; k0 < F_; k0 += 4) {
    v2f a  = *(const v2f*)(arow + k0);
    v2f bb = *(const v2f*)(brow + k0);
    c = __builtin_amdgcn_wmma_f32_16x16x4_f32(false, a, false, bb,
                                              (short)0, c, false, false);
  }

  const int mbase = (lane >> 4) << 3;          // 0 or 8
  float* out = Wh + ((size_t)(b * N_ + mt * 16)) * F_ + gt * 16 + r;
#pragma unroll
  for (int g = 0; g < 8; ++g) out[(size_t)(mbase + g) * F_] = c[g];
}

// ---------------------------------------------------------------------------
// Kernel 2: u[bn] = Wh[bn,:]·a[0:128], v[bn] = Wh[bn,:]·a[128:256]
// ---------------------------------------------------------------------------
__global__ __launch_bounds__(256) void uv_kernel(
    const float* __restrict__ Wh, const float* __restrict__ a,
    float* __restrict__ u, float* __restrict__ v) {
  const int n = blockIdx.x * 256 + threadIdx.x;   // 0 .. B*N-1
  const float* row = Wh + (size_t)n * F_;
  float su = 0.f, sv = 0.f;
#pragma unroll 4
  for (int k = 0; k < F_; k += 4) {
    float4 x  = *(const float4*)(row + k);
    float4 a1 = *(const float4*)(a + k);
    float4 a2 = *(const float4*)(a + F_ + k);
    su += x.x * a1.x + x.y * a1.y + x.z * a1.z + x.w * a1.w;
    sv += x.x * a2.x + x.y * a2.y + x.z * a2.z + x.w * a2.w;
  }
  u[n] = su;
  v[n] = sv;
}

// ---------------------------------------------------------------------------
// Kernel 3 (one block per batch): vmax = max_j v, w_j = exp(v_j - vmax),
// colMean[f] = mean_j Wh[b,j,f] (fallback for fully-masked softmax rows).
// ---------------------------------------------------------------------------
__global__ __launch_bounds__(256) void stats_kernel(
    const float* __restrict__ Wh, const float* __restrict__ v,
    float* __restrict__ w, float* __restrict__ colMean) {
  __shared__ float red[256];
  const int b = blockIdx.x, t = threadIdx.x;
  const float* vb = v + b * N_;
  float m = -3.0e38f;
  for (int i = t; i < N_; i += 256) m = fmaxf(m, vb[i]);
  red[t] = m;
  __syncthreads();
  for (int s = 128; s > 0; s >>= 1) {
    if (t < s) red[t] = fmaxf(red[t], red[t + s]);
    __syncthreads();
  }
  const float vm = red[0];
  for (int i = t; i < N_; i += 256) w[b * N_ + i] = expf(vb[i] - vm);
  if (t < F_) {
    const float* base = Wh + (size_t)b * N_ * F_ + t;
    float s = 0.f;
    for (int j = 0; j < N_; ++j) s += base[(size_t)j * F_];
    colMean[b * F_ + t] = s * (1.0f / (float)N_);
  }
}

// ---------------------------------------------------------------------------
// Kernel 4: out[b,i,f] = elu( sum_j P_ij * Wh[b,j,f] / Z_i ),
//   P_ij = (u_i + v_j > 0) ? exp(v_j - vmax) : 0,  Z_i = sum_j P_ij,
//   Z_i == 0  ->  uniform attention  ->  colMean.
// Block = (batch, 16-row i-tile); 8 waves cover the 8 f-tiles.
// Wh staged into LDS in 64-row chunks; P built in registers; K=4 f32 WMMA.
// ---------------------------------------------------------------------------
#define CJ 64
__global__ __launch_bounds__(256) void attn_kernel(
    const float* __restrict__ Wh, const float* __restrict__ u,
    const float* __restrict__ v, const float* __restrict__ w,
    const float* __restrict__ colMean, float* __restrict__ out) {
  __shared__ float WhS[CJ][F_];
  __shared__ float vS[CJ];
  __shared__ float wS[CJ];

  const int b    = blockIdx.x >> 7;
  const int it   = blockIdx.x & 127;
  const int t    = threadIdx.x;
  const int lane = t & 31;
  const int warp = t >> 5;                    // f-tile index 0..7
  const int r    = lane & 15;
  const int kp   = (lane >> 4) << 1;          // 0 or 2

  const float u_i = u[b * N_ + it * 16 + r];  // A-frag row for this lane
  const float* Whb = Wh + (size_t)b * N_ * F_;
  const float* vb  = v + b * N_;
  const float* wb  = w + b * N_;

  v8f c = {};
  float zp = 0.f;

  for (int j0 = 0; j0 < N_; j0 += CJ) {
    __syncthreads();                          // protect LDS reuse
    {
      const float4* src = (const float4*)(Whb + (size_t)j0 * F_);
      float4* dst = (float4*)(&WhS[0][0]);
#pragma unroll
      for (int q = 0; q < (CJ * F_ / 4) / 256; ++q) {
        const int idx = t + q * 256;
        dst[idx] = src[idx];
      }
      if (t < CJ) {
        vS[t] = vb[j0 + t];
        wS[t] = wb[j0 + t];
      }
      if (j0 + CJ < N_)                       // global_prefetch_b8 next chunk
        __builtin_prefetch(Whb + (size_t)(j0 + CJ) * F_ + t * 4, 0, 1);
    }
    __syncthreads();

#pragma unroll
    for (int ks = 0; ks < CJ / 4; ++ks) {
      const int jj = ks * 4 + kp;
      const float p0 = (u_i + vS[jj]     > 0.f) ? wS[jj]     : 0.f;
      const float p1 = (u_i + vS[jj + 1] > 0.f) ? wS[jj + 1] : 0.f;
      zp += p0 + p1;
      v2f a  = {p0, p1};
      v2f bb = {WhS[jj][warp * 16 + r], WhS[jj + 1][warp * 16 + r]};
      c = __builtin_amdgcn_wmma_f32_16x16x4_f32(false, a, false, bb,
                                                (short)0, c, false, false);
    }
  }

  // lanes r and r+16 hold the two K-halves of row r's Z partial
  zp += __shfl_xor(zp, 16, 32);               // now lane L holds Z[L%16]

  const int mbase = (lane >> 4) << 3;         // 0 or 8
  const float cm = colMean[b * F_ + warp * 16 + r];
  float* op = out + ((size_t)(b * N_ + it * 16)) * F_ + warp * 16 + r;
#pragma unroll
  for (int g = 0; g < 8; ++g) {
    const int irow = mbase + g;
    const float Z  = __shfl(zp, irow, 32);    // lane irow holds Z[irow]
    float val = (Z > 0.f) ? (c[g] / Z) : cm;  // Z==0: uniform softmax == mean
    val = (val > 0.f) ? val : (expf(val) - NEG_ELU_ONE);
    op[(size_t)irow * F_] = val;
  }
}

// ---------------------------------------------------------------------------
extern "C" void kernel_launch(void* const* d_in, const int* in_sizes, int n_in,
                              void* d_out, int out_size, void* d_ws,
                              size_t ws_size, hipStream_t stream) {
  const float* h = (const float*)d_in[0];   // [8,2048,128]
  const float* W = (const float*)d_in[1];   // [128,128]
  const float* a = (const float*)d_in[2];   // [256,1]
  // d_in[3] (A1) and d_in[4] (A2) feed only dead code in the reference:
  // the returned output never reads `adj`/`hid`, so we skip ~1.1 TFLOP.
  (void)in_sizes; (void)n_in; (void)out_size; (void)ws_size;

  float* Wh      = (float*)d_ws;            // 8*2048*128 = 2,097,152 floats
  float* u       = Wh + (size_t)B_ * N_ * F_;   // 16384
  float* v       = u + B_ * N_;                 // 16384
  float* w       = v + B_ * N_;                 // 16384
  float* colMean = w + B_ * N_;                 // 1024
  float* out     = (float*)d_out;               // [8,2048,128] fp32

  wh_gemm_kernel<<<1024, 256, 0, stream>>>(h, W, Wh);
  uv_kernel<<<(B_ * N_) / 256, 256, 0, stream>>>(Wh, a, u, v);
  stats_kernel<<<B_, 256, 0, stream>>>(Wh, v, w, colMean);
  attn_kernel<<<B_ * (N_ / 16), 256, 0, stream>>>(Wh, u, v, w, colMean, out);
}